// ODEFuncNN3Layer_17867063951492
// MI455X (gfx1250) — compile-verified
//
#include <hip/hip_runtime.h>
#include <math.h>

// out = tanh(y @ W1.T + b1) @ W2.T + b2
//   y: [B,2] f32, W1: [50,2], b1: [50], W2: [2,50], b2: [2], out: [B,2]
// Layer 1 computed as H^T tiles via V_WMMA_F32_16X16X4_F32 (W1 = constant A
// operand, batch tile = B operand, bias preloaded in C). tanh via hardware
// v_tanh_f32. Layer 2 (50->2) as per-lane FMAs + one cross-half shuffle
// reduction. Each wave processes 32 batch rows per iteration with fully
// coalesced, unpredicated 256B loads/stores (no exec-mask branching).

typedef float v2f __attribute__((ext_vector_type(2)));
typedef float v8f __attribute__((ext_vector_type(8)));

#define HID 50
#define DIM 2

#if __has_builtin(__builtin_amdgcn_tanhf)
#  define FAST_TANH(x) __builtin_amdgcn_tanhf(x)
#elif __has_builtin(__builtin_amdgcn_tanh_f32)
#  define FAST_TANH(x) __builtin_amdgcn_tanh_f32(x)
#else
#  define FAST_TANH(x) tanhf(x)
#endif

__device__ __forceinline__ void process_pair(
    int pair, int nrows, bool guard,
    const float* __restrict__ y, float* __restrict__ out,
    const v2f* a1, const v8f* c1,
    const float (*w2r0)[8], const float (*w2r1)[8],
    float bias0, float bias1, int lane, int half)
{
    constexpr int vlim[4] = {8, 8, 8, 2};
    const int row = pair * 32 + lane;   // lanes 0-15: tile A, lanes 16-31: tile B

    // One coalesced 256B load per wave covers both 16-row tiles.
    v2f yv = {0.0f, 0.0f};
    if (!guard || row < nrows) yv = *(const v2f*)(y + (size_t)row * DIM);

    // Tile B's rows live in the upper half-wave; swap them down.
    v2f ysw;
    ysw.x = __shfl_xor(yv.x, 16);
    ysw.y = __shfl_xor(yv.y, 16);

    // B operand layout (4x16 f32): lanes 0-15 carry K=0,1; lanes 16-31 carry
    // K=2,3 which are zero padding (DIM=2).
    const bool lo = (half == 0);
    v2f bA = { lo ? yv.x  : 0.0f, lo ? yv.y  : 0.0f };
    v2f bB = { lo ? ysw.x : 0.0f, lo ? ysw.y : 0.0f };

    // Layer 1: 8 WMMAs -> H^T for both tiles, bias folded into C.
    v8f accA[4], accB[4];
    #pragma unroll
    for (int t = 0; t < 4; ++t) {
        accA[t] = __builtin_amdgcn_wmma_f32_16x16x4_f32(
            false, a1[t], false, bA, (short)0, c1[t], false, false);
        accB[t] = __builtin_amdgcn_wmma_f32_16x16x4_f32(
            false, a1[t], false, bB, (short)0, c1[t], false, false);
    }

    // tanh + layer-2 partial dot products (hidden index j split across the
    // lane pair l, l^16).
    float sA0 = 0.0f, sA1 = 0.0f, sB0 = 0.0f, sB1 = 0.0f;
    #pragma unroll
    for (int t = 0; t < 4; ++t) {
        #pragma unroll
        for (int v = 0; v < 8; ++v) {
            if (v < vlim[t]) {
                float hA = FAST_TANH(accA[t][v]);
                float hB = FAST_TANH(accB[t][v]);
                sA0 = fmaf(hA, w2r0[t][v], sA0);
                sA1 = fmaf(hA, w2r1[t][v], sA1);
                sB0 = fmaf(hB, w2r0[t][v], sB0);
                sB1 = fmaf(hB, w2r1[t][v], sB1);
            }
        }
    }
    // Cross-half reduction completes the 50-term sums; afterwards every lane
    // holds the full result for its batch column.
    sA0 += __shfl_xor(sA0, 16);
    sA1 += __shfl_xor(sA1, 16);
    sB0 += __shfl_xor(sB0, 16);
    sB1 += __shfl_xor(sB1, 16);

    // Lanes 0-15 store tile A rows, lanes 16-31 store tile B rows:
    // row = pair*32 + lane for all lanes -> one coalesced 256B store.
    float o0 = (lo ? sA0 : sB0) + bias0;
    float o1 = (lo ? sA1 : sB1) + bias1;
    if (!guard || row < nrows) {
        v2f o = { o0, o1 };
        *(v2f*)(out + (size_t)row * DIM) = o;
    }
}

__global__ __launch_bounds__(256) void odefunc_wmma_kernel(
    const float* __restrict__ y,
    const float* __restrict__ W1,
    const float* __restrict__ b1,
    const float* __restrict__ W2,
    const float* __restrict__ b2,
    float* __restrict__ out,
    int nrows)
{
    const int lane = threadIdx.x & 31;
    const int half = lane >> 4;
    const int col  = lane & 15;
    constexpr int vlim[4] = {8, 8, 8, 2};

    // ---------------- per-wave constant preload (amortized) ----------------
    // A tiles (16x4 f32 layout): lanes 0-15 hold M=lane with VGPR0=K0,
    // VGPR1=K1 (= W1[j][0..1]); lanes 16-31 hold K=2,3 (zero padding).
    v2f a1[4];
    // C tiles: bias b1[j]; C/D layout: lane l, VGPR v -> j = 16t+v+8*(l/16).
    v8f c1[4];
    // W2 staged per lane to match the D-layout element j = 16t+v+8*half.
    float w2r0[4][8];
    float w2r1[4][8];

    #pragma unroll
    for (int t = 0; t < 4; ++t) {
        v2f a = {0.0f, 0.0f};
        int j = 16 * t + col;
        if (half == 0 && j < HID) {
            a.x = W1[j * DIM + 0];
            a.y = W1[j * DIM + 1];
        }
        a1[t] = a;
        #pragma unroll
        for (int v = 0; v < 8; ++v) {
            int jj = 16 * t + v + 8 * half;
            c1[t][v] = (jj < HID) ? b1[jj] : 0.0f;
            float w0 = 0.0f, w1 = 0.0f;
            if (v < vlim[t] && jj < HID) {
                w0 = W2[0 * HID + jj];
                w1 = W2[1 * HID + jj];
            }
            w2r0[t][v] = w0;
            w2r1[t][v] = w1;
        }
    }
    const float bias0 = b2[0];
    const float bias1 = b2[1];

    const int waveId = (int)((blockIdx.x * blockDim.x + threadIdx.x) >> 5);
    const int nWaves = (int)((gridDim.x * blockDim.x) >> 5);

    // Guard-free main loop over full 32-row pairs (B=2M is divisible by 32).
    const int nFull = nrows >> 5;
    for (int pair = waveId; pair < nFull; pair += nWaves) {
        process_pair(pair, nrows, /*guard=*/false, y, out,
                     a1, c1, w2r0, w2r1, bias0, bias1, lane, half);
    }
    // Single guarded tail iteration (at most one wave executes it).
    if ((nrows & 31) != 0 && (nFull % nWaves) == waveId) {
        process_pair(nFull, nrows, /*guard=*/true, y, out,
                     a1, c1, w2r0, w2r1, bias0, bias1, lane, half);
    }
}

extern "C" void kernel_launch(void* const* d_in, const int* in_sizes, int n_in,
                              void* d_out, int out_size, void* d_ws, size_t ws_size,
                              hipStream_t stream) {
    // setup_inputs order: t, y, W1, b1, W2, b2
    const float* y  = (const float*)d_in[1];
    const float* W1 = (const float*)d_in[2];
    const float* b1 = (const float*)d_in[3];
    const float* W2 = (const float*)d_in[4];
    const float* b2 = (const float*)d_in[5];
    float* out = (float*)d_out;

    const int nrows = in_sizes[1] / DIM;   // 2,000,000

    // 1024 blocks x 8 wave32 = 8192 waves; ~7.6 iterations per wave over
    // 62.5k row-pairs — saturates HBM while amortizing the weight preload.
    const int block = 256;
    const int grid  = 1024;
    odefunc_wmma_kernel<<<grid, block, 0, stream>>>(y, W1, b1, W2, b2, out, nrows);
}